// ObjectDetectionModelV2_15187004359199
// MI455X (gfx1250) — compile-verified
//
#include <hip/hip_runtime.h>
#include <cstdint>
#include <cstddef>

// ---------------------------------------------------------------------------
// Problem constants (from reference): B=8, C=256, H=W=128, K=20
// ---------------------------------------------------------------------------
#define BB   8
#define CC   256
#define HH   128
#define WW   128
#define KK   20
#define HWW  (HH * WW)          // 16384 pixels per (b, channel) plane
#define NSEG (BB * KK)          // 160 segments
#define CHUNK 1024              // pixels per block in segmax kernel (4 per thread)
#define LTAB_STRIDE 257         // 20 x 257 padded LDS table -> bank = (seg+c)%64

typedef float v2f __attribute__((ext_vector_type(2)));
typedef float v4f __attribute__((ext_vector_type(4)));   // native vector: OK for nontemporal builtins
typedef float v8f __attribute__((ext_vector_type(8)));

// Order-preserving float <-> uint mapping: enables exact integer atomicMax.
__device__ __forceinline__ unsigned encOrd(float f) {
    unsigned u = __float_as_uint(f);
    return (u & 0x80000000u) ? ~u : (u | 0x80000000u);
}
__device__ __forceinline__ float decOrd(unsigned u) {
    return __uint_as_float((u & 0x80000000u) ? (u & 0x7FFFFFFFu) : ~u);
}
// encOrd(-inf) == 0x007FFFFF : init value so empty segments decode to -inf -> relu 0
#define ENC_NEG_INF 0x007FFFFFu

// Non-temporal b128 helpers for the single-use HBM streams (TH=NT: don't
// pollute L2 -- encoded is 134 MB read exactly once).
__device__ __forceinline__ v4f ntload4(const float* p) {
    return __builtin_nontemporal_load((const v4f*)p);
}
__device__ __forceinline__ void ntstore4(float* p, v4f v) {
    __builtin_nontemporal_store(v, (v4f*)p);
}

// ---------------------------------------------------------------------------
// Kernel 0: init global segment-max table to encoded(-inf)
// ---------------------------------------------------------------------------
__global__ __launch_bounds__(256) void k_init(unsigned* __restrict__ tab) {
    int i = blockIdx.x * 256 + threadIdx.x;
    if (i < NSEG * CC) tab[i] = ENC_NEG_INF;
}

// ---------------------------------------------------------------------------
// Kernel 1: fused argmax + class passthrough copy + segment max.
// Grid: B * (HW/CHUNK) = 8*16 = 128 blocks x 256 threads; thread owns 4
// consecutive pixels so every bulk access is a coalesced b128 op.
// Segment ids stay in registers between phases (same pixel ownership), so
// LDS is used only for the 20x257 padded max table (ds_max_u32).
// Single pass over `encoded` (134 MB) and `encoded_classes` (10.5 MB r + w).
// ---------------------------------------------------------------------------
__global__ __launch_bounds__(256) void k_segmax(const float* __restrict__ enc,
                                                const float* __restrict__ ec,
                                                unsigned* __restrict__ tab,
                                                float* __restrict__ outc) {
    const int blk = blockIdx.x;
    const int b = blk >> 4;                 // 16 chunks per batch image
    const int pixbase = (blk & 15) << 10;   // * CHUNK
    const int tid = threadIdx.x;
    const int p4 = tid << 2;                // first of this thread's 4 pixels

    __shared__ unsigned ltab[KK * LTAB_STRIDE];     // ~20.5 KB

    for (int i = tid; i < KK * LTAB_STRIDE; i += 256) ltab[i] = 0u;  // 0 < ENC_NEG_INF: harmless
    __syncthreads();

    // Phase 1: per-pixel argmax over K classes (4 pixels/thread, b128);
    // simultaneously stream the classes tensor to the output copy.
    int bk0 = 0, bk1 = 0, bk2 = 0, bk3 = 0;
    {
        const size_t off = (size_t)b * KK * HWW + (size_t)(pixbase + p4);
        const float* src = ec + off;
        float* dst = outc + off;
        v4f best = ntload4(src);
        ntstore4(dst, best);
        for (int k = 1; k < KK; ++k) {
            v4f v = ntload4(src + (size_t)k * HWW);
            ntstore4(dst + (size_t)k * HWW, v);
            if (v.x > best.x) { best.x = v.x; bk0 = k; }  // strict '>': first occurrence (jnp.argmax)
            if (v.y > best.y) { best.y = v.y; bk1 = k; }
            if (v.z > best.z) { best.z = v.z; bk2 = k; }
            if (v.w > best.w) { best.w = v.w; bk3 = k; }
        }
    }

    // Segment row bases for this thread's 4 pixels (registers, no LDS).
    const int sb0 = bk0 * LTAB_STRIDE;
    const int sb1 = bk1 * LTAB_STRIDE;
    const int sb2 = bk2 * LTAB_STRIDE;
    const int sb3 = bk3 * LTAB_STRIDE;

    // Phase 2: stream all 256 channels (b128, non-temporal), LDS atomic max.
    const float* base = enc + ((size_t)b * CC) * HWW + pixbase + p4;
    #pragma unroll 2
    for (int c = 0; c < CC; ++c) {
        v4f v = ntload4(base + (size_t)c * HWW);
        atomicMax(&ltab[sb0 + c], encOrd(v.x));
        atomicMax(&ltab[sb1 + c], encOrd(v.y));
        atomicMax(&ltab[sb2 + c], encOrd(v.z));
        atomicMax(&ltab[sb3 + c], encOrd(v.w));
    }
    __syncthreads();

    // Phase 3: merge block-local table into global table.
    for (int i = tid; i < KK * CC; i += 256) {
        const int s = i >> 8;       // / 256
        const int c = i & (CC - 1);
        atomicMax(&tab[(size_t)(b * KK + s) * CC + c], ltab[s * LTAB_STRIDE + c]);
    }
}

// ---------------------------------------------------------------------------
// Kernel 2: hid = relu(vec) @ w1^T + b1 via V_WMMA_F32_16X16X4_F32.
// M=160 (10 tiles) x N=128 (8 tiles), K=256 (64 WMMA steps of K=4).
// One wave32 block per (Mtile, Ntile); EXEC all ones as WMMA requires.
// A layout (16x4 f32): lane%16 = M row; lane/16 selects K-pair {0,1} vs {2,3};
// v2f = (K, K+1). B (4x16) mirrors with lane%16 = N column.
// D (16x16 f32, v8f): vgpr r, lane L -> M = r + (L/16)*8, N = L%16.
// ---------------------------------------------------------------------------
__global__ __launch_bounds__(32) void k_mlp1(const unsigned* __restrict__ tab,
                                             const float* __restrict__ w1,
                                             const float* __restrict__ b1,
                                             float* __restrict__ hid) {
    const int lane = threadIdx.x;
    const int l16 = lane & 15;
    const int kh  = lane >> 4;
    const int m0 = blockIdx.x << 4;
    const int n0 = blockIdx.y << 4;

    const unsigned* vrow = tab + (size_t)(m0 + l16) * CC;   // A: vec row m
    const float*    wrow = w1  + (size_t)(n0 + l16) * CC;   // B: w1 row n (B[k][n] = w1[n][k])

    v8f acc = {};
    for (int kk = 0; kk < CC; kk += 4) {
        const int ki = kk + (kh << 1);
        float a0 = decOrd(vrow[ki]);
        float a1 = decOrd(vrow[ki + 1]);
        v2f a    = { a0 > 0.0f ? a0 : 0.0f, a1 > 0.0f ? a1 : 0.0f };  // relu(segmax)
        v2f bmat = { wrow[ki], wrow[ki + 1] };
        acc = __builtin_amdgcn_wmma_f32_16x16x4_f32(
            /*neg_a=*/false, a, /*neg_b=*/false, bmat,
            /*c_mod=*/(short)0, acc, /*reuse_a=*/false, /*reuse_b=*/false);
    }

    const int n = n0 + l16;
    const float bias = b1[n];
    #pragma unroll
    for (int r = 0; r < 8; ++r) {
        const int m = m0 + r + (kh << 3);
        hid[(size_t)m * 128 + n] = acc[r] + bias;
    }
}

// ---------------------------------------------------------------------------
// Kernel 3: bbox = sigmoid(hid @ w2^T + b2); keep classes k>=1. 640 dots.
// ---------------------------------------------------------------------------
__global__ __launch_bounds__(256) void k_mlp2(const float* __restrict__ hid,
                                              const float* __restrict__ w2,
                                              const float* __restrict__ b2,
                                              float* __restrict__ out) {
    const int i = blockIdx.x * 256 + threadIdx.x;
    if (i >= NSEG * 4) return;
    const int m = i >> 2;
    const int p = i & 3;
    const float* h = hid + (size_t)m * 128;
    const float* w = w2  + (size_t)p * 128;
    float s = b2[p];
    #pragma unroll 4
    for (int o = 0; o < 128; ++o) s += h[o] * w[o];
    const float sg = 1.0f / (1.0f + expf(-s));
    const int b = m / KK;
    const int k = m % KK;
    if (k >= 1) out[(size_t)(b * (KK - 1) + (k - 1)) * 4 + p] = sg;
}

// ---------------------------------------------------------------------------
// Launch
// ---------------------------------------------------------------------------
extern "C" void kernel_launch(void* const* d_in, const int* in_sizes, int n_in,
                              void* d_out, int out_size, void* d_ws, size_t ws_size,
                              hipStream_t stream) {
    const float* enc = (const float*)d_in[0];   // encoded         (8,256,128,128)
    const float* ec  = (const float*)d_in[1];   // encoded_classes (8,20,128,128)
    const float* w1  = (const float*)d_in[2];   // (128,256)
    const float* b1  = (const float*)d_in[3];   // (128,)
    const float* w2  = (const float*)d_in[4];   // (4,128)
    const float* b2  = (const float*)d_in[5];   // (4,)

    float* out = (float*)d_out;                 // [608 bbox] ++ [classes copy]
    float* outc = out + (size_t)BB * (KK - 1) * 4;   // 2432 B offset: 16B-aligned

    unsigned* tab = (unsigned*)d_ws;                                   // 160*256 u32 = 160 KB
    float* hid = (float*)((char*)d_ws + (size_t)NSEG * CC * sizeof(unsigned)); // 160*128 f32 = 80 KB

    k_init  <<<NSEG, 256, 0, stream>>>(tab);
    k_segmax<<<BB * (HWW / CHUNK), 256, 0, stream>>>(enc, ec, tab, outc);
    k_mlp1  <<<dim3(NSEG / 16, 128 / 16), 32, 0, stream>>>(tab, w1, b1, hid);
    k_mlp2  <<<(NSEG * 4 + 255) / 256, 256, 0, stream>>>(hid, w2, b2, out);
}